// Preprocessor_41429254537723
// MI455X (gfx1250) — compile-verified
//
#include <hip/hip_runtime.h>
#include <cstdint>
#include <cstddef>

// ---------------------------------------------------------------------------
// Feature preprocessor for MI455X (gfx1250, wave32).
// Memory-bound streaming transform: ~203MB traffic -> ~8.7us at 23.3 TB/s.
// Column tiles of 256 are segment-homogeneous (all boundaries are multiples
// of 256), so blockIdx.x selects the transform with a scalar branch.
// ---------------------------------------------------------------------------

#define B_ROWS   8192
#define F_IN     2624
#define F_OUT    3584
#define O0       512      // end of binary
#define O1       1024     // end of logistic
#define O2       2048     // end of continuous
#define O3       2560     // end of quantile (input); enum starts here
#define NQV      11       // quantile boundaries per column
#define E_VALS   16
#define MISSING_F (-1388135708.0f)
#define MINF_C   (-6.0f)
#define MAXF_C   (6.0f)
#define ROWS_PER_BLOCK 16

typedef int v4i __attribute__((ext_vector_type(4)));
typedef __attribute__((address_space(1))) v4i* gptr_v4i;
typedef __attribute__((address_space(3))) v4i* lptr_v4i;

__global__ __launch_bounds__(256) void preprocessor_kernel(
    const float* __restrict__ x,
    const float* __restrict__ means,
    const float* __restrict__ stddevs,
    const float* __restrict__ qb,          // [NQ, 11]
    const float* __restrict__ enum_values, // [NE, 16]
    float* __restrict__ out)
{
    const int tid = threadIdx.x;
    const int c0  = blockIdx.x << 8;                 // input-column tile base
    const int r0  = blockIdx.y * ROWS_PER_BLOCK;

    if (c0 < O0) {
        // ------------------------- binary: (x != 0) * not_missing ----------
        const int c = c0 + tid;
        #pragma unroll
        for (int rr = 0; rr < ROWS_PER_BLOCK; ++rr) {
            const size_t r = (size_t)(r0 + rr);
            const float v = x[r * F_IN + c];
            out[r * F_OUT + c] = (v != 0.0f && v != MISSING_F) ? 1.0f : 0.0f;
        }
    } else if (c0 < O1) {
        // ------------------------- logistic: -log(1/clip(x)-1) * nm --------
        const int c = c0 + tid;
        #pragma unroll
        for (int rr = 0; rr < ROWS_PER_BLOCK; ++rr) {
            const size_t r = (size_t)(r0 + rr);
            const float v  = x[r * F_IN + c];
            const float nm = (v != MISSING_F) ? 1.0f : 0.0f;
            const float cp = fminf(fmaxf(v, 0.01f), 0.99f);
            float o = -logf(1.0f / cp - 1.0f) * nm;
            out[r * F_OUT + c] = fminf(fmaxf(o, MINF_C), MAXF_C);
        }
    } else if (c0 < O2) {
        // ------------------------- continuous: clip((x-mu)/sd) * nm --------
        const int c = c0 + tid;
        const int j = c - O1;
        const float mu = means[j];
        const float sd = stddevs[j];
        #pragma unroll
        for (int rr = 0; rr < ROWS_PER_BLOCK; ++rr) {
            const size_t r = (size_t)(r0 + rr);
            const float v  = x[r * F_IN + c];
            const float nm = (v != MISSING_F) ? 1.0f : 0.0f;
            const float o  = fminf(fmaxf((v - mu) / sd, MINF_C), MAXF_C) * nm;
            out[r * F_OUT + c] = o;
        }
    } else if (c0 < O3) {
        // ------------------------- quantile bucketization ------------------
        // Stage this tile's 256x11 boundary table into LDS via the CDNA5
        // async global->LDS DMA path, then lift each thread's 11 boundaries
        // into registers for the row loop.
        __shared__ __align__(16) float sqb[256 * NQV];   // 11264 bytes
        const int   j0   = c0 - O2;
        const float* gsrc = qb + (size_t)j0 * NQV;       // 16B-aligned slice
        constexpr int NV4 = (256 * NQV) / 4;             // 704 float4 chunks
#if __has_builtin(__builtin_amdgcn_global_load_async_to_lds_b128)
        for (int i = tid; i < NV4; i += 256) {
            const void* g = (const void*)(gsrc + (size_t)i * 4);
            void*       l = (void*)(&sqb[i * 4]);
            __builtin_amdgcn_global_load_async_to_lds_b128(
                (gptr_v4i)g, (lptr_v4i)l, 0, 0);
        }
#if __has_builtin(__builtin_amdgcn_s_wait_asynccnt)
        __builtin_amdgcn_s_wait_asynccnt(0);
#else
        asm volatile("s_wait_asynccnt 0" ::: "memory");
#endif
        __syncthreads();
#else
        for (int i = tid; i < NV4; i += 256)
            ((float4*)sqb)[i] = ((const float4*)gsrc)[i];
        __syncthreads();
#endif
        float q[NQV];
        #pragma unroll
        for (int k = 0; k < NQV; ++k) q[k] = sqb[tid * NQV + k];

        const int c = c0 + tid;
        #pragma unroll 4
        for (int rr = 0; rr < ROWS_PER_BLOCK; ++rr) {
            const size_t r = (size_t)(r0 + rr);
            const float v  = x[r * F_IN + c];
            const float nm = (v != MISSING_F) ? 1.0f : 0.0f;
            float cnt = 0.0f, left = -1e20f, right = 1e20f;
            #pragma unroll
            for (int k = 0; k < NQV; ++k) {
                const bool ge = (v >= q[k]);
                cnt  += ge ? 1.0f : 0.0f;
                left  = ge ? fmaxf(left, q[k]) : left;    // max over ge set
                right = ge ? right : fminf(right, q[k]);  // min over lt set
            }
            const float interp =
                ((cnt - 1.0f) + (v - left) / (right - left + 1e-6f)) / 10.0f;
            float o = (v >= q[NQV - 1]) ? 1.0f
                    : ((v <= q[0]) ? 0.0f : interp);
            o *= nm;
            out[r * F_OUT + c] = fminf(fmaxf(o, MINF_C), MAXF_C);
        }
    } else {
        // ------------------------- enum one-hot (64 cols -> 1024 cols) -----
        // 4 threads per input column; each thread emits one float4 (16B),
        // so consecutive lanes store consecutive 16B chunks (b128 stores).
        const int k       = tid >> 2;     // enum column 0..63
        const int quarter = tid & 3;      // which 4 of the 16 one-hot slots
        const int c       = O3 + k;
        const float4 ev   = ((const float4*)enum_values)[k * 4 + quarter];
        const int ocol    = O3 + k * E_VALS + quarter * 4;
        #pragma unroll
        for (int rr = 0; rr < ROWS_PER_BLOCK; ++rr) {
            const size_t r = (size_t)(r0 + rr);
            const float v  = x[r * F_IN + c];
            const float nm = (v != MISSING_F) ? 1.0f : 0.0f;
            float4 o;
            o.x = (v == ev.x) ? nm : 0.0f;
            o.y = (v == ev.y) ? nm : 0.0f;
            o.z = (v == ev.z) ? nm : 0.0f;
            o.w = (v == ev.w) ? nm : 0.0f;
            *(float4*)(out + r * F_OUT + ocol) = o;
        }
    }
}

extern "C" void kernel_launch(void* const* d_in, const int* in_sizes, int n_in,
                              void* d_out, int out_size, void* d_ws, size_t ws_size,
                              hipStream_t stream) {
    const float* x       = (const float*)d_in[0];
    const float* means   = (const float*)d_in[1];
    const float* stddevs = (const float*)d_in[2];
    const float* qb      = (const float*)d_in[3];
    const float* enum_v  = (const float*)d_in[4];
    float* out           = (float*)d_out;

    const int rows = in_sizes[0] / F_IN;               // 8192
    dim3 grid((F_IN + 255) / 256,                      // 11 column tiles
              rows / ROWS_PER_BLOCK);                  // 512 row tiles
    preprocessor_kernel<<<grid, 256, 0, stream>>>(x, means, stddevs, qb,
                                                  enum_v, out);
}